// DAGLayer_39324720562471
// MI455X (gfx1250) — compile-verified
//
#include <hip/hip_runtime.h>

typedef float v2f __attribute__((ext_vector_type(2)));
typedef float v8f __attribute__((ext_vector_type(8)));

#define CW        64          // channel width
#define HW        1024        // H*W
#define BATCH     64
#define TP        32          // spatial positions per workgroup tile
#define ROWSTRIDE 66          // padded LDS row stride (floats): 64 ch + 2 pad
#define AREA      (TP * ROWSTRIDE)
#define NMAT      16          // 2 pre + 14 edge matrices

// ---------------------------------------------------------------------------
// Pack weights into WMMA A-fragment order:
//   frag(m, w, k, lane) = { W[m][16w + (lane&15)][4k + 2*(lane>>4)],
//                           W[m][16w + (lane&15)][4k + 2*(lane>>4) + 1] }
// so the main kernel issues one coalesced b64 load per A fragment.
// ---------------------------------------------------------------------------
__global__ void pack_weights(const float* __restrict__ Wpre,
                             const float* __restrict__ Wedge,
                             float* __restrict__ pw) {
    int t = blockIdx.x * blockDim.x + threadIdx.x;   // one thread per float2
    if (t >= NMAT * 4 * 16 * 32) return;
    int lane = t & 31;
    int k    = (t >> 5) & 15;
    int w    = (t >> 9) & 3;
    int m    = t >> 11;
    const float* Wm = (m < 2) ? (Wpre + m * CW * CW) : (Wedge + (m - 2) * CW * CW);
    int o  = 16 * w + (lane & 15);
    int c0 = 4 * k + 2 * (lane >> 4);
    float2 v = make_float2(Wm[o * CW + c0], Wm[o * CW + c0 + 1]);
    ((float2*)pw)[t] = v;
}

// ---------------------------------------------------------------------------
// Fused DAG kernel: one workgroup = one batch element x 32 spatial positions.
// All 6 DAG states live in LDS; 16 GEMMs via V_WMMA_F32_16X16X4_F32.
// ---------------------------------------------------------------------------
__global__ __launch_bounds__(128)
void dag_kernel(const float* __restrict__ x0, const float* __restrict__ x1,
                const float* __restrict__ pw, float* __restrict__ out) {
    __shared__ float lds[7 * AREA];     // states 0..5 + x staging buffer

    const int b    = blockIdx.x >> 5;   // 32 tiles per batch
    const int tile = blockIdx.x & 31;
    const int p0   = tile * TP;
    const int lane = threadIdx.x & 31;
    const int wv   = threadIdx.x >> 5;  // wave id: owns output channels 16wv..16wv+15
    const int lpos = lane & 15;         // N (position) index within tile
    const int lhi  = lane >> 4;         // K/M half-select

    float* xbuf = lds + 6 * AREA;

    // Stage a [64ch x 32pos] input tile into LDS (pos-major, padded rows).
    auto stage = [&](const float* __restrict__ xp) {
        #pragma unroll
        for (int i = 0; i < 4; ++i) {
            int f   = (i * 128 + threadIdx.x) * 4;        // flat float index
            int ch  = f >> 5;                             // 32 floats per channel row
            int pos = f & 31;
            const float4 v = *(const float4*)(xp + (size_t)b * CW * HW +
                                              (size_t)ch * HW + p0 + pos);
            xbuf[(pos + 0) * ROWSTRIDE + ch] = v.x;
            xbuf[(pos + 1) * ROWSTRIDE + ch] = v.y;
            xbuf[(pos + 2) * ROWSTRIDE + ch] = v.z;
            xbuf[(pos + 3) * ROWSTRIDE + ch] = v.w;
        }
    };

    // acc[pt] += W[m] x (relu?)src  for this wave's 16-row output slice.
    auto conv_acc = [&](v8f acc[2], const float* __restrict__ src, int m, bool relu) {
        const v2f* __restrict__ wfrag =
            ((const v2f*)pw) + (size_t)((m * 4 + wv) * 16) * 32 + lane;
        #pragma unroll
        for (int k = 0; k < 16; ++k) {
            v2f a = wfrag[k * 32];                        // A fragment (b64, coalesced)
            #pragma unroll
            for (int pt = 0; pt < 2; ++pt) {
                const float* bp = src + (pt * 16 + lpos) * ROWSTRIDE + 4 * k + 2 * lhi;
                v2f bv = *(const v2f*)bp;                 // B fragment (ds_load_b64)
                if (relu) { bv.x = fmaxf(bv.x, 0.0f); bv.y = fmaxf(bv.y, 0.0f); }
                acc[pt] = __builtin_amdgcn_wmma_f32_16x16x4_f32(
                    /*neg_a=*/false, a, /*neg_b=*/false, bv,
                    /*c_mod=*/(short)0, acc[pt], /*reuse_a=*/false, /*reuse_b=*/false);
            }
        }
    };

    // Write a node state to LDS (for downstream edges) and optionally to HBM.
    auto store_state = [&](v8f acc[2], float* __restrict__ dst, float* __restrict__ gout) {
        #pragma unroll
        for (int pt = 0; pt < 2; ++pt) {
            float* d = dst + (pt * 16 + lpos) * ROWSTRIDE + 16 * wv + 8 * lhi;
            #pragma unroll
            for (int v = 0; v < 4; ++v) {
                v2f pr; pr.x = acc[pt][2 * v]; pr.y = acc[pt][2 * v + 1];
                *(v2f*)(d + 2 * v) = pr;                  // ds_store_b64
            }
            if (gout) {
                #pragma unroll
                for (int v = 0; v < 8; ++v) {
                    int ch = 16 * wv + 8 * lhi + v;
                    gout[(size_t)ch * HW + p0 + pt * 16 + lpos] = acc[pt][v];
                }
            }
        }
    };

    float* S[6];
    #pragma unroll
    for (int i = 0; i < 6; ++i) S[i] = lds + i * AREA;

    // ---- preprocess: states 0,1 = W_pre[i] x x_i (no ReLU) ----
    stage(x0);
    __syncthreads();
    {
        v8f acc[2] = {v8f{}, v8f{}};
        conv_acc(acc, xbuf, 0, false);
        __syncthreads();                 // all waves done reading xbuf
        store_state(acc, S[0], nullptr);
    }
    stage(x1);
    __syncthreads();                     // xbuf=x1 ready, S0 committed
    {
        v8f acc[2] = {v8f{}, v8f{}};
        conv_acc(acc, xbuf, 1, false);
        store_state(acc, S[1], nullptr);
    }
    __syncthreads();                     // S1 committed

    // ---- DAG nodes: sum over edges = continued WMMA accumulation ----
    int e = 0;
    for (int n = 0; n < 4; ++n) {
        v8f acc[2] = {v8f{}, v8f{}};
        const int ns = 2 + n;
        for (int s = 0; s < ns; ++s)
            conv_acc(acc, S[s], 2 + e + s, true);         // ReLU fused into B load
        e += ns;
        store_state(acc, S[2 + n],
                    out + (size_t)b * (4 * CW * HW) + (size_t)n * CW * HW);
        __syncthreads();                 // node state visible to next node
    }
}

// ---------------------------------------------------------------------------
extern "C" void kernel_launch(void* const* d_in, const int* in_sizes, int n_in,
                              void* d_out, int out_size, void* d_ws, size_t ws_size,
                              hipStream_t stream) {
    const float* x0    = (const float*)d_in[0];
    const float* x1    = (const float*)d_in[1];
    const float* Wpre  = (const float*)d_in[2];
    const float* Wedge = (const float*)d_in[3];
    float*       pw    = (float*)d_ws;      // 16 * 64 * 64 floats = 256 KB packed

    const int packN = NMAT * 4 * 16 * 32;   // one thread per packed float2
    pack_weights<<<(packN + 255) / 256, 256, 0, stream>>>(Wpre, Wedge, pw);

    dag_kernel<<<BATCH * (HW / TP), 128, 0, stream>>>(x0, x1, pw, (float*)d_out);
}